// DeformAttn_65773129171137
// MI455X (gfx1250) — compile-verified
//
#include <hip/hip_runtime.h>
#include <hip/hip_bf16.h>

// Problem constants (match reference)
#define NB  8
#define LQn 2048
#define CC  256
#define MM  8
#define LLv 4
#define PPn 4
#define DHn 32
#define SSn 3840   // 2048+1024+512+256

typedef __attribute__((ext_vector_type(2))) float v2f;
typedef __attribute__((ext_vector_type(8))) float v8f;

#define TRS 4   // 64-row strips (4 row-tiles share each B fragment)

// ---------------------------------------------------------------------------
// Wave-level fp32 WMMA strip: one wave computes a (16*TRS) x 16 strip of
// D = A @ W with V_WMMA_F32_16X16X4_F32. K and LDB are compile-time constants
// so every fragment load lowers to an immediate-offset global load (no per-
// iteration 64-bit address math) and the unrolled windows form load clauses
// that run ahead of the WMMAs.
//
// A-fragment (ISA 7.12.2, 32-bit A 16x4): lanes 0-15 hold row=lane, K={k,k+1};
// lanes 16-31 hold row=lane-16, K={k+2,k+3}  -> aligned float2 (b64) load.
// B-fragment (4x16, rows striped across lanes): lanes 0-15 col=lane K={k,k+1};
// lanes 16-31 col=lane-16, K={k+2,k+3}.
// D element (i,lane): row = row0 + i + hi*8, col = col0 + (lane&15).
// ---------------------------------------------------------------------------
template <int K, int LDB>
__device__ __forceinline__ void wmma_strip_f32(const float* __restrict__ A,
                                               const float* __restrict__ W,
                                               int row0, int col0,
                                               v8f* __restrict__ c) {
    const int lane = threadIdx.x & 31;
    const int hi   = lane >> 4;
    const int lm   = lane & 15;
    const float* __restrict__ arow[TRS];
#pragma unroll
    for (int t = 0; t < TRS; ++t)
        arow[t] = A + (row0 + 16 * t + lm) * K + hi * 2;
    const float* __restrict__ bcol = W + (hi * 2) * LDB + col0 + lm;
#pragma unroll 8
    for (int k = 0; k < K; k += 4) {
        v2f b;
        b[0] = bcol[k * LDB];            // immediate-offset global_load_b32
        b[1] = bcol[(k + 1) * LDB];
#pragma unroll
        for (int t = 0; t < TRS; ++t) {
            float2 af = *(const float2*)(arow[t] + k);   // global_load_b64
            v2f a; a[0] = af.x; a[1] = af.y;
            c[t] = __builtin_amdgcn_wmma_f32_16x16x4_f32(
                       /*neg_a=*/false, a, /*neg_b=*/false, b,
                       /*c_mod=*/(short)0, c[t],
                       /*reuse_a=*/false, /*reuse_b=*/false);
        }
    }
}

// ---------------------------------------------------------------------------
// Kernel 1/5: plain GEMM + bias  (value projection, output projection)
// ---------------------------------------------------------------------------
template <int K, int COUT>
__global__ void gemm_bias_kernel(const float* __restrict__ A,
                                 const float* __restrict__ W,
                                 const float* __restrict__ bias,
                                 float* __restrict__ D,
                                 int R) {
    const int tilesN = COUT >> 4;
    const int strip = blockIdx.x * (blockDim.x >> 5) + (threadIdx.x >> 5);
    if (strip >= (R / (16 * TRS)) * tilesN) return;
    const int row0 = (strip / tilesN) * (16 * TRS);
    const int col0 = (strip % tilesN) << 4;
    v8f c[TRS] = {};
    wmma_strip_f32<K, COUT>(A, W, row0, col0, c);
    const int lane = threadIdx.x & 31;
    const int hi = lane >> 4, lm = lane & 15;
    const int col = col0 + lm;
    const float bv = bias[col];
#pragma unroll
    for (int t = 0; t < TRS; ++t)
#pragma unroll
        for (int i = 0; i < 8; ++i) {
            const int r = row0 + 16 * t + i + hi * 8;
            D[(size_t)r * COUT + col] = c[t][i] + bv;
        }
}

// ---------------------------------------------------------------------------
// Kernel 2: offset GEMM fused with loc computation
// loc = ref[n,q,l] + (q@W_off + b_off)[col] / norm[l],  col = m*16 + l*4 + p
// ---------------------------------------------------------------------------
__global__ void gemm_loc_kernel(const float* __restrict__ Q,
                                const float* __restrict__ Woff,
                                const float* __restrict__ boff,
                                const float* __restrict__ ref,   // (N*LQ, L)
                                float* __restrict__ loc) {       // (N*LQ, 128)
    constexpr int COUT = MM * LLv * PPn;             // 128
    const int tilesN = COUT >> 4;                    // 8
    const int strip = blockIdx.x * (blockDim.x >> 5) + (threadIdx.x >> 5);
    if (strip >= ((NB * LQn) / (16 * TRS)) * tilesN) return;
    const int row0 = (strip / tilesN) * (16 * TRS);
    const int col0 = (strip % tilesN) << 4;
    v8f c[TRS] = {};
    wmma_strip_f32<CC, COUT>(Q, Woff, row0, col0, c);
    const int lane = threadIdx.x & 31;
    const int hi = lane >> 4, lm = lane & 15;
    const int col = col0 + lm;
    const int l = (col >> 2) & 3;
    const float invNorm[LLv] = {1.f/2048.f, 1.f/1024.f, 1.f/512.f, 1.f/256.f};
    const float bv = boff[col];
#pragma unroll
    for (int t = 0; t < TRS; ++t)
#pragma unroll
        for (int i = 0; i < 8; ++i) {
            const int r = row0 + 16 * t + i + hi * 8;
            const float off = c[t][i] + bv;
            loc[(size_t)r * COUT + col] = ref[(size_t)r * LLv + l] + off * invNorm[l];
        }
}

// ---------------------------------------------------------------------------
// Kernel 3: attention GEMM fused with softmax over L*P=16 (== one 16-col tile)
// Row values live across one 16-lane half -> shfl_xor(width=16) reductions.
// ---------------------------------------------------------------------------
__global__ void gemm_softmax_kernel(const float* __restrict__ Q,
                                    const float* __restrict__ Wat,
                                    const float* __restrict__ bat,
                                    float* __restrict__ attn) {  // (N*LQ, 128)
    constexpr int COUT = MM * LLv * PPn;             // 128
    const int tilesN = COUT >> 4;                    // 8, col0 = m*16
    const int strip = blockIdx.x * (blockDim.x >> 5) + (threadIdx.x >> 5);
    if (strip >= ((NB * LQn) / (16 * TRS)) * tilesN) return;
    const int row0 = (strip / tilesN) * (16 * TRS);
    const int col0 = (strip % tilesN) << 4;
    v8f c[TRS] = {};
    wmma_strip_f32<CC, COUT>(Q, Wat, row0, col0, c);
    const int lane = threadIdx.x & 31;
    const int hi = lane >> 4, lm = lane & 15;
    const int col = col0 + lm;
    const float bv = bat[col];
#pragma unroll
    for (int t = 0; t < TRS; ++t)
#pragma unroll
        for (int i = 0; i < 8; ++i) {
            float v = c[t][i] + bv;
            float mx = v;
#pragma unroll
            for (int s = 8; s >= 1; s >>= 1)
                mx = fmaxf(mx, __shfl_xor(mx, s, 16));
            float e = __expf(v - mx);
            float sm = e;
#pragma unroll
            for (int s = 8; s >= 1; s >>= 1)
                sm += __shfl_xor(sm, s, 16);
            const int r = row0 + 16 * t + i + hi * 8;
            attn[(size_t)r * COUT + col] = e / sm;
        }
}

// ---------------------------------------------------------------------------
// Kernel 4: deformable gather. One wave per (n,q,m); lane = channel (DH=32).
// value layout (N,S,M,DH) contiguous == (N*S, C) rows -> 128B coalesced loads,
// value (31.5 MB) fits in 192MB L2 so re-reads stay on-chip.
// ---------------------------------------------------------------------------
__global__ void deform_kernel(const float* __restrict__ value, // (N*S, C)
                              const float* __restrict__ loc,   // (N*LQ, 128)
                              const float* __restrict__ attn,  // (N*LQ, 128)
                              float* __restrict__ out) {       // (N*LQ, C)
    const int lane = threadIdx.x & 31;
    const int idx = blockIdx.x * (blockDim.x >> 5) + (threadIdx.x >> 5);
    if (idx >= NB * LQn * MM) return;
    const int m = idx & (MM - 1);
    const int row = idx >> 3;            // n*LQ + q
    const int n = row >> 11;             // / LQ

    const int lens[LLv]   = {2048, 1024, 512, 256};
    const int starts[LLv] = {0, 2048, 3072, 3584};

    // coalesced fetch of 16 loc (lanes 0-15) + 16 attn (lanes 16-31) scalars
    const size_t sbase = (size_t)row * (MM * LLv * PPn) + m * 16;
    float ld = (lane < 16) ? loc[sbase + lane] : attn[sbase + (lane - 16)];

    const float* __restrict__ vbase =
        value + ((size_t)n * SSn) * CC + m * DHn + lane;

    float acc = 0.f;
#pragma unroll
    for (int j = 0; j < LLv * PPn; ++j) {
        const int l = j >> 2;
        const int T = lens[l];
        const float lv = __shfl(ld, j, 32);
        const float av = __shfl(ld, 16 + j, 32);
        const float pos = lv * (float)T - 0.5f;
        const float x0f = floorf(pos);
        const float frac = pos - x0f;
        const int x0 = (int)x0f;
        const float w0 = (1.f - frac) * ((x0 >= 0 && x0 < T) ? 1.f : 0.f) * av;
        const float w1 = frac * ((x0 + 1 >= 0 && x0 + 1 < T) ? 1.f : 0.f) * av;
        const int i0 = min(max(x0, 0), T - 1);
        const int i1 = min(max(x0 + 1, 0), T - 1);
        const float* __restrict__ vl = vbase + (size_t)starts[l] * CC;
        acc += w0 * vl[(size_t)i0 * CC] + w1 * vl[(size_t)i1 * CC];
    }
    out[(size_t)row * CC + m * DHn + lane] = acc;
}

// ---------------------------------------------------------------------------
extern "C" void kernel_launch(void* const* d_in, const int* in_sizes, int n_in,
                              void* d_out, int out_size, void* d_ws, size_t ws_size,
                              hipStream_t stream) {
    const float* query  = (const float*)d_in[0];
    const float* refpt  = (const float*)d_in[1];
    const float* inflat = (const float*)d_in[2];
    // d_in[3]/d_in[4] = temporal_lens / level_start_index (static, hardcoded)
    const float* W_off  = (const float*)d_in[5];
    const float* b_off  = (const float*)d_in[6];
    const float* W_attn = (const float*)d_in[7];
    const float* b_attn = (const float*)d_in[8];
    const float* W_val  = (const float*)d_in[9];
    const float* b_val  = (const float*)d_in[10];
    const float* W_out  = (const float*)d_in[11];
    const float* b_out  = (const float*)d_in[12];

    float* out_main = (float*)d_out;                                // (N,LQ,C)
    float* out_loc  = out_main + (size_t)NB * LQn * CC;             // (N,LQ,M,L,P)
    float* out_attn = out_loc + (size_t)NB * LQn * MM * LLv * PPn;  // (N,LQ,M,L,P)

    float* ws_value  = (float*)d_ws;                                // (N*S, C)
    float* ws_deform = ws_value + (size_t)NB * SSn * CC;            // (N*LQ, C)

    const int threads = 256;            // 8 waves/block (wave32)
    const int wpb = threads >> 5;

    // 1) value = input_flatten @ W_val + b_val   (30720x256x256)
    {
        const int strips = ((NB * SSn) / (16 * TRS)) * (CC >> 4);   // 7680
        gemm_bias_kernel<CC, CC><<<strips / wpb, threads, 0, stream>>>(
            inflat, W_val, b_val, ws_value, NB * SSn);
    }
    // 2) loc = ref + (query @ W_off + b_off) / norm
    {
        const int strips = ((NB * LQn) / (16 * TRS)) * ((MM * LLv * PPn) >> 4); // 2048
        gemm_loc_kernel<<<strips / wpb, threads, 0, stream>>>(
            query, W_off, b_off, refpt, out_loc);
    }
    // 3) attn = softmax(query @ W_attn + b_attn)
    {
        const int strips = ((NB * LQn) / (16 * TRS)) * ((MM * LLv * PPn) >> 4); // 2048
        gemm_softmax_kernel<<<strips / wpb, threads, 0, stream>>>(
            query, W_attn, b_attn, out_attn);
    }
    // 4) deformable gather -> ws_deform
    {
        const int waves = NB * LQn * MM;                            // 131072
        deform_kernel<<<waves / wpb, threads, 0, stream>>>(
            ws_value, out_loc, out_attn, ws_deform);
    }
    // 5) out = deform @ W_out + b_out   (16384x256x256)
    {
        const int strips = ((NB * LQn) / (16 * TRS)) * (CC >> 4);   // 4096
        gemm_bias_kernel<CC, CC><<<strips / wpb, threads, 0, stream>>>(
            ws_deform, W_out, b_out, out_main, NB * LQn);
    }
}